// GNN_42502996362010
// MI455X (gfx1250) — compile-verified
//
#include <hip/hip_runtime.h>
#include <hip/hip_bf16.h>

typedef __attribute__((ext_vector_type(2))) float v2f;
typedef __attribute__((ext_vector_type(8))) float v8f;

#define NU 100000
#define NI 100000
#define DD 128
#define EE 800000
#define NPART 512

// ---------------------------------------------------------------------------
// GEMM: y = x @ W^T + b    (x:[n,128], W:[128,128] row-major, y:[n,128])
// One block = one 16-row tile. 8 waves, wave w owns output cols [16w,16w+16).
// K-loop: 32 x v_wmma_f32_16x16x4_f32, A-frags from LDS, B-frags from global.
// ---------------------------------------------------------------------------
__global__ __launch_bounds__(256) void gemm_xwT_wmma(
    const float* __restrict__ x, const float* __restrict__ w,
    const float* __restrict__ bias, float* __restrict__ y, int nrows)
{
    __shared__ float xt[16 * 128];   // 8 KB x-tile
    const int tid  = threadIdx.x;
    const int row0 = blockIdx.x * 16;
    if (row0 >= nrows) return;       // uniform per block

    // cooperative, coalesced load of the 16x128 tile
#pragma unroll
    for (int i = 0; i < 8; ++i) {
        int idx = tid + i * 256;
        xt[idx] = x[(size_t)row0 * 128 + idx];
    }
    __syncthreads();

    const int wv   = tid >> 5;        // wave id = col tile 0..7
    const int lane = tid & 31;
    const int hi   = lane >> 4;       // half: lanes 0-15 -> K {k,k+1}, 16-31 -> {k+2,k+3}
    const int r    = lane & 15;       // A: row M=r ; B: col N=r
    const int col0 = wv * 16;
    const float* wrow = w + (size_t)(col0 + r) * 128;  // B[k][N=r] = W[r_col][k]

    v8f c = {};
#pragma unroll
    for (int k0 = 0; k0 < 128; k0 += 4) {
        const int ka = k0 + 2 * hi;
        v2f a; a.x = xt[r * 128 + ka]; a.y = xt[r * 128 + ka + 1];
        v2f b; b.x = wrow[ka];         b.y = wrow[ka + 1];
        c = __builtin_amdgcn_wmma_f32_16x16x4_f32(
                /*neg_a=*/false, a, /*neg_b=*/false, b,
                /*c_mod=*/(short)0, c, /*reuse_a=*/false, /*reuse_b=*/false);
    }

    const float bj = bias[col0 + r];
    // D layout: VGPR v -> M = v + 8*half, N = r
#pragma unroll
    for (int vv = 0; vv < 8; ++vv) {
        int m = vv + 8 * hi;
        y[(size_t)(row0 + m) * 128 + col0 + r] = c[vv] + bj;
    }
}

// ---------------------------------------------------------------------------
// Small utility kernels
// ---------------------------------------------------------------------------
__global__ void fill_f32(float* __restrict__ p, float val, int n)
{
    int i = blockIdx.x * blockDim.x + threadIdx.x;
    if (i < n) p[i] = val;
}

__global__ void count_edges(const int* __restrict__ dst, int e_cnt,
                            float* __restrict__ counts)
{
    int e = blockIdx.x * blockDim.x + threadIdx.x;
    if (e < e_cnt) atomicAdd(&counts[dst[e]], 1.0f);
}

// partial[b][d] = sum over node chunk of counts[n]*k[n][d]   (deterministic 2-pass)
__global__ __launch_bounds__(128) void ksum_partial(
    const float* __restrict__ k, const float* __restrict__ counts, int n,
    float* __restrict__ partial)
{
    const int d = threadIdx.x;
    const int chunk = (n + gridDim.x - 1) / gridDim.x;
    const int n0 = blockIdx.x * chunk;
    const int n1 = min(n, n0 + chunk);
    float acc = 0.f;
    for (int i = n0; i < n1; ++i) acc += counts[i] * k[(size_t)i * 128 + d];
    partial[blockIdx.x * 128 + d] = acc;
}

__global__ __launch_bounds__(128) void ksum_final(
    const float* __restrict__ partial, int nb, float* __restrict__ ksum)
{
    const int d = threadIdx.x;
    float acc = 0.f;
    for (int b = 0; b < nb; ++b) acc += partial[b * 128 + d];
    ksum[d] = acc;
}

// out[i] = invscale * sum_j W[i][j]*v[j]   (W row-major 128x128)
__global__ __launch_bounds__(128) void matvec128(
    const float* __restrict__ W, const float* __restrict__ v,
    float* __restrict__ out, float invscale)
{
    const int i = threadIdx.x;
    float acc = 0.f;
    for (int j = 0; j < 128; ++j) acc += W[i * 128 + j] * v[j];
    out[i] = acc * invscale;
}

// per-node: t[n] = q[n].wk (wk already has 1/sqrt(D)), sdot[n] = q[n].k[n]/sqrt(D)
__global__ __launch_bounds__(256) void node_scores(
    const float* __restrict__ q, const float* __restrict__ k,
    const float* __restrict__ wk, float* __restrict__ t,
    float* __restrict__ sdot, int n, float invscale)
{
    const int lane = threadIdx.x & 31;
    const int node = (blockIdx.x * blockDim.x + threadIdx.x) >> 5;
    if (node >= n) return;
    const float4 qv = ((const float4*)(q + (size_t)node * 128))[lane];
    const float4 kv = ((const float4*)(k + (size_t)node * 128))[lane];
    const float4 wv = ((const float4*)wk)[lane];
    float ta = qv.x * wv.x + qv.y * wv.y + qv.z * wv.z + qv.w * wv.w;
    float sa = qv.x * kv.x + qv.y * kv.y + qv.z * kv.z + qv.w * kv.w;
#pragma unroll
    for (int off = 16; off > 0; off >>= 1) {
        ta += __shfl_down(ta, off);
        sa += __shfl_down(sa, off);
    }
    if (lane == 0) { t[node] = ta; sdot[node] = sa * invscale; }
}

__device__ __forceinline__ void smx_combine(float& m, float& s, float m2, float s2)
{
    float mm = fmaxf(m, m2);
    float acc = 0.f;
    if (s  > 0.f) acc += s  * expf(m  - mm);
    if (s2 > 0.f) acc += s2 * expf(m2 - mm);
    m = mm; s = acc;
}

__global__ __launch_bounds__(256) void softmax_partial(
    const float* __restrict__ x, int n, float2* __restrict__ partial)
{
    __shared__ float sm[256];
    __shared__ float ss[256];
    const int chunk = (n + gridDim.x - 1) / gridDim.x;
    const int n0 = blockIdx.x * chunk;
    const int n1 = min(n, n0 + chunk);
    float m = -__builtin_inff(), s = 0.f;
    for (int i = n0 + threadIdx.x; i < n1; i += 256) {
        float v = x[i];
        if (v > m) { s = s * expf(m - v) + 1.f; m = v; }
        else       { s += expf(v - m); }
    }
    sm[threadIdx.x] = m; ss[threadIdx.x] = s;
    __syncthreads();
    for (int st = 128; st > 0; st >>= 1) {
        if (threadIdx.x < st) {
            float mm = sm[threadIdx.x], s1 = ss[threadIdx.x];
            smx_combine(mm, s1, sm[threadIdx.x + st], ss[threadIdx.x + st]);
            sm[threadIdx.x] = mm; ss[threadIdx.x] = s1;
        }
        __syncthreads();
    }
    if (threadIdx.x == 0) partial[blockIdx.x] = make_float2(sm[0], ss[0]);
}

__global__ void softmax_final(const float2* __restrict__ partial, int nb,
                              float* __restrict__ MS)
{
    if (threadIdx.x == 0 && blockIdx.x == 0) {
        float m = -__builtin_inff(), s = 0.f;
        for (int i = 0; i < nb; ++i) smx_combine(m, s, partial[i].x, partial[i].y);
        MS[0] = m; MS[1] = s;
    }
}

// order-independent float atomic max via int/uint monotone mapping
__device__ __forceinline__ void atomicMaxF(float* addr, float val)
{
    if (val >= 0.f) atomicMax((int*)addr, __float_as_int(val));
    else atomicMin((unsigned int*)addr, (unsigned int)__float_as_int(val));
}

__global__ void edge_max(const int* __restrict__ src, const int* __restrict__ dst,
                         int e_cnt, const float* __restrict__ t,
                         float* __restrict__ m)
{
    int e = blockIdx.x * blockDim.x + threadIdx.x;
    if (e < e_cnt) atomicMaxF(&m[dst[e]], t[src[e]]);
}

__global__ void edge_sumexp(const int* __restrict__ src, const int* __restrict__ dst,
                            int e_cnt, const float* __restrict__ t,
                            const float* __restrict__ m, float* __restrict__ ssum)
{
    int e = blockIdx.x * blockDim.x + threadIdx.x;
    if (e < e_cnt) atomicAdd(&ssum[dst[e]], expf(t[src[e]] - m[dst[e]]));
}

// one wave per edge: acc[dst] += alpha * v[src]
__global__ __launch_bounds__(256) void edge_scatter(
    const int* __restrict__ src, const int* __restrict__ dst, int e_cnt,
    const float* __restrict__ t, const float* __restrict__ m,
    const float* __restrict__ ssum, const float* __restrict__ v,
    float* __restrict__ acc)
{
    const int lane = threadIdx.x & 31;
    const int e = (blockIdx.x * blockDim.x + threadIdx.x) >> 5;
    if (e >= e_cnt) return;
    const int s = src[e], d = dst[e];
    const float alpha = expf(t[s] - m[d]) / ssum[d];
    const float4 vv = ((const float4*)(v + (size_t)s * 128))[lane];
    float* out = acc + (size_t)d * 128 + lane * 4;
    atomicAdd(out + 0, alpha * vv.x);
    atomicAdd(out + 1, alpha * vv.y);
    atomicAdd(out + 2, alpha * vv.z);
    atomicAdd(out + 3, alpha * vv.w);
}

// acc[n][d] += (exp(sdot[n]-M)/S) * v[n][d]
__global__ void add_self(float* __restrict__ acc, const float* __restrict__ v,
                         const float* __restrict__ sdot,
                         const float* __restrict__ MS, int n)
{
    size_t i = (size_t)blockIdx.x * blockDim.x + threadIdx.x;
    if (i >= (size_t)n * 128) return;
    size_t node = i >> 7;
    float a = expf(sdot[node] - MS[0]) / MS[1];
    acc[i] += a * v[i];
}

// ---------------------------------------------------------------------------
// Host side
// ---------------------------------------------------------------------------
struct LayerP {
    const float *qw_u, *qb_u, *kw_u, *kb_u, *vw_u, *vb_u;
    const float *qw_i, *qb_i, *kw_i, *kb_i, *vw_i, *vb_i;
    const float *W_u2i, *W_i2u;
};

struct Ws {
    float *q_u, *k_u, *v_u, *q_i, *k_i, *v_i;
    float *counts_u, *counts_i, *t_u, *t_i, *sdot_u, *sdot_i;
    float *m_u, *m_i, *ssum_u, *ssum_i;
    float *ksum_u, *ksum_i, *wk_u2i, *wk_i2u, *MS_u, *MS_i, *partial;
};

static void run_layer(hipStream_t s, const LayerP& P, const Ws& W,
                      const float* xu, const float* xi,
                      const int* u2i_src, const int* u2i_dst,
                      const int* i2u_src, const int* i2u_dst,
                      float* acc_u, float* acc_i)
{
    const float invscale = 1.0f / sqrtf((float)DD);
    const dim3 b256(256);

    hipMemsetAsync(acc_u, 0, (size_t)NU * DD * sizeof(float), s);
    hipMemsetAsync(acc_i, 0, (size_t)NI * DD * sizeof(float), s);
    hipMemsetAsync(W.counts_u, 0, NU * sizeof(float), s);
    hipMemsetAsync(W.counts_i, 0, NI * sizeof(float), s);
    hipMemsetAsync(W.ssum_u, 0, NU * sizeof(float), s);
    hipMemsetAsync(W.ssum_i, 0, NI * sizeof(float), s);
    fill_f32<<<(NU + 255) / 256, b256, 0, s>>>(W.m_u, -__builtin_inff(), NU);
    fill_f32<<<(NI + 255) / 256, b256, 0, s>>>(W.m_i, -__builtin_inff(), NI);

    // 6 WMMA GEMMs
    gemm_xwT_wmma<<<NU / 16, b256, 0, s>>>(xu, P.qw_u, P.qb_u, W.q_u, NU);
    gemm_xwT_wmma<<<NU / 16, b256, 0, s>>>(xu, P.kw_u, P.kb_u, W.k_u, NU);
    gemm_xwT_wmma<<<NU / 16, b256, 0, s>>>(xu, P.vw_u, P.vb_u, W.v_u, NU);
    gemm_xwT_wmma<<<NI / 16, b256, 0, s>>>(xi, P.qw_i, P.qb_i, W.q_i, NI);
    gemm_xwT_wmma<<<NI / 16, b256, 0, s>>>(xi, P.kw_i, P.kb_i, W.k_i, NI);
    gemm_xwT_wmma<<<NI / 16, b256, 0, s>>>(xi, P.vw_i, P.vb_i, W.v_i, NI);

    // degree counts per destination
    count_edges<<<(EE + 255) / 256, b256, 0, s>>>(i2u_dst, EE, W.counts_u);
    count_edges<<<(EE + 255) / 256, b256, 0, s>>>(u2i_dst, EE, W.counts_i);

    // ksum = counts @ k (deterministic two-pass)
    ksum_partial<<<NPART, 128, 0, s>>>(W.k_u, W.counts_u, NU, W.partial);
    ksum_final<<<1, 128, 0, s>>>(W.partial, NPART, W.ksum_u);
    ksum_partial<<<NPART, 128, 0, s>>>(W.k_i, W.counts_i, NI, W.partial);
    ksum_final<<<1, 128, 0, s>>>(W.partial, NPART, W.ksum_i);

    // wk = (W @ ksum) / sqrt(D)
    matvec128<<<1, 128, 0, s>>>(P.W_u2i, W.ksum_i, W.wk_u2i, invscale);
    matvec128<<<1, 128, 0, s>>>(P.W_i2u, W.ksum_u, W.wk_i2u, invscale);

    // per-node logits t (edge) and sdot (self)
    node_scores<<<(NU * 32 + 255) / 256, b256, 0, s>>>(W.q_u, W.k_u, W.wk_u2i,
                                                       W.t_u, W.sdot_u, NU, invscale);
    node_scores<<<(NI * 32 + 255) / 256, b256, 0, s>>>(W.q_i, W.k_i, W.wk_i2u,
                                                       W.t_i, W.sdot_i, NI, invscale);

    // self softmax (over all nodes)
    softmax_partial<<<NPART, b256, 0, s>>>(W.sdot_u, NU, (float2*)W.partial);
    softmax_final<<<1, 32, 0, s>>>((float2*)W.partial, NPART, W.MS_u);
    softmax_partial<<<NPART, b256, 0, s>>>(W.sdot_i, NI, (float2*)W.partial);
    softmax_final<<<1, 32, 0, s>>>((float2*)W.partial, NPART, W.MS_i);

    // segment softmax over edges (dst segments)
    edge_max<<<(EE + 255) / 256, b256, 0, s>>>(i2u_src, i2u_dst, EE, W.t_i, W.m_u);
    edge_max<<<(EE + 255) / 256, b256, 0, s>>>(u2i_src, u2i_dst, EE, W.t_u, W.m_i);
    edge_sumexp<<<(EE + 255) / 256, b256, 0, s>>>(i2u_src, i2u_dst, EE, W.t_i, W.m_u, W.ssum_u);
    edge_sumexp<<<(EE + 255) / 256, b256, 0, s>>>(u2i_src, u2i_dst, EE, W.t_u, W.m_i, W.ssum_i);

    // alpha-weighted scatter: one wave per edge
    edge_scatter<<<(EE * 32 + 255) / 256, b256, 0, s>>>(i2u_src, i2u_dst, EE,
        W.t_i, W.m_u, W.ssum_u, W.v_i, acc_u);
    edge_scatter<<<(EE * 32 + 255) / 256, b256, 0, s>>>(u2i_src, u2i_dst, EE,
        W.t_u, W.m_i, W.ssum_i, W.v_u, acc_i);

    // add self-attention message
    add_self<<<(int)(((size_t)NU * DD + 255) / 256), b256, 0, s>>>(acc_u, W.v_u, W.sdot_u, W.MS_u, NU);
    add_self<<<(int)(((size_t)NI * DD + 255) / 256), b256, 0, s>>>(acc_i, W.v_i, W.sdot_i, W.MS_i, NI);
}

extern "C" void kernel_launch(void* const* d_in, const int* in_sizes, int n_in,
                              void* d_out, int out_size, void* d_ws, size_t ws_size,
                              hipStream_t stream)
{
    const float* x_user = (const float*)d_in[0];
    const float* x_item = (const float*)d_in[1];

    auto getLayer = [&](int base) {
        LayerP p;
        p.qw_u = (const float*)d_in[base + 0];  p.qb_u = (const float*)d_in[base + 1];
        p.kw_u = (const float*)d_in[base + 2];  p.kb_u = (const float*)d_in[base + 3];
        p.vw_u = (const float*)d_in[base + 4];  p.vb_u = (const float*)d_in[base + 5];
        p.qw_i = (const float*)d_in[base + 6];  p.qb_i = (const float*)d_in[base + 7];
        p.kw_i = (const float*)d_in[base + 8];  p.kb_i = (const float*)d_in[base + 9];
        p.vw_i = (const float*)d_in[base + 10]; p.vb_i = (const float*)d_in[base + 11];
        p.W_u2i = (const float*)d_in[base + 12];
        p.W_i2u = (const float*)d_in[base + 13];
        return p;
    };
    const LayerP L1 = getLayer(2);
    const LayerP L2 = getLayer(16);

    const int* ei_u2i = (const int*)d_in[30];
    const int* ei_i2u = (const int*)d_in[31];
    const int* u2i_src = ei_u2i;       const int* u2i_dst = ei_u2i + EE;
    const int* i2u_src = ei_i2u;       const int* i2u_dst = ei_i2u + EE;

    // carve workspace
    float* p = (float*)d_ws;
    auto take = [&](size_t n) { float* r = p; p += n; return r; };
    const size_t ND = (size_t)NU * DD;

    Ws W;
    W.q_u = take(ND); W.k_u = take(ND); W.v_u = take(ND);
    W.q_i = take(ND); W.k_i = take(ND); W.v_i = take(ND);
    float* xu1 = take(ND);
    float* xi1 = take(ND);
    W.counts_u = take(NU); W.counts_i = take(NI);
    W.t_u = take(NU); W.t_i = take(NI);
    W.sdot_u = take(NU); W.sdot_i = take(NI);
    W.m_u = take(NU); W.m_i = take(NI);
    W.ssum_u = take(NU); W.ssum_i = take(NI);
    W.ksum_u = take(128); W.ksum_i = take(128);
    W.wk_u2i = take(128); W.wk_i2u = take(128);
    W.MS_u = take(2); W.MS_i = take(2);
    W.partial = take((size_t)NPART * 128);

    float* out_u = (float*)d_out;
    float* out_i = out_u + ND;

    // layer 1: inputs -> xu1/xi1 ; layer 2: xu1/xi1 -> d_out
    run_layer(stream, L1, W, x_user, x_item, u2i_src, u2i_dst, i2u_src, i2u_dst, xu1, xi1);
    run_layer(stream, L2, W, xu1, xi1, u2i_src, u2i_dst, i2u_src, i2u_dst, out_u, out_i);
}